// FConv2d_57071525430227
// MI455X (gfx1250) — compile-verified
//
#include <hip/hip_runtime.h>

typedef float v2f __attribute__((ext_vector_type(2)));
typedef float v8f __attribute__((ext_vector_type(8)));

#define BB   16
#define CIN  128
#define COUT 256
#define LL   32

#define QP   33          // padded K dim for weights, odd -> conflict-free banks
#define CP   155         // padded wrapped-channel dim for x slab, odd -> conflict-free

// Block: 256 threads = 8 waves, handles (b, row y, half-row x0).
// Wave wid owns the d-pair {wid, 16-wid} (wave 0: {0,8}); for wid>0 both C tiles share
// one B fragment: B = X[(8*d0-q)%128] + X[(8*d1-q)%128] (mirror-channel symmetry).
// K axis is traversed as m = 31-q in BOTH operands (reduction order is free), so every
// lane's A and B loads share one base with ascending 8-bit ds_load_2addr immediates.
__global__ __launch_bounds__(256) void fconv2d_wmma_kernel(
    const float* __restrict__ x,
    const float* __restrict__ w,
    float* __restrict__ out)
{
    // sW2[tap][n][m(pad 33)] = 0.5 * w[n][31-m][2-dy][2-dx]  : 19.0 KB
    __shared__ float sW2[9 * 16 * QP];
    // sX2[row 0..2][xi 0..17][slot c in 1..153 (pad 155)], slot c holds channel (c+95)&127 : 33.5 KB
    __shared__ float sX2[3 * 18 * CP];

    const int tid = threadIdx.x;
    const int bid = blockIdx.x;          // 1024 blocks
    const int b   = bid >> 6;
    const int rem = bid & 63;
    const int y   = rem >> 1;
    const int x0  = (rem & 1) << 4;

    // --- stage flipped+halved weights, K-major, q-axis reversed (m = 31-q) ---
    for (int idx = tid; idx < 16 * 32 * 9; idx += 256) {
        int n = idx / 288;
        int r = idx % 288;
        int q = r / 9;
        int t = r % 9;
        int i = t / 3, j = t % 3;
        sW2[((2 - i) * 3 + (2 - j)) * (16 * QP) + n * QP + (31 - q)] = 0.5f * w[idx];
    }
    // --- stage x slab, wrapped channel axis innermost ---
    for (int idx = tid; idx < 153 * 54; idx += 256) {
        int c   = 1 + idx / 54;          // 1..153
        int r2  = idx % 54;
        int row = r2 / 18;
        int xi  = r2 % 18;
        int ch  = (c + 95) & 127;        // slot c holds channel c-33 mod 128
        int gy  = y + row;
        int gx  = x0 + xi;
        float v = 0.0f;
        if (gy < LL && gx < LL)
            v = x[((b * CIN + ch) * LL + gy) * LL + gx];
        sX2[(row * 18 + xi) * CP + c] = v;
    }
    __syncthreads();

    const int lane = tid & 31;
    const int wid  = tid >> 5;
    const int col  = lane & 15;     // pixel col for B/C, filter row for A
    const int hi   = lane >> 4;     // K-half: m offset 0 or 2

    const int d0 = wid;
    const int d1 = wid ? (16 - wid) : 8;

    v8f c0 = {0.f, 0.f, 0.f, 0.f, 0.f, 0.f, 0.f, 0.f};
    v8f c1 = {0.f, 0.f, 0.f, 0.f, 0.f, 0.f, 0.f, 0.f};

    if (wid != 0) {
        for (int dy = 0; dy < 3; ++dy) {
            for (int dx = 0; dx < 3; ++dx) {
                const float* ap  = &sW2[(dy * 3 + dx) * (16 * QP) + col * QP + 2 * hi];
                const int    sb  = (dy * 18 + col + dx) * CP + 2 + 2 * hi;
                const float* b0p = &sX2[sb + 8 * d0];
                const float* b1p = &sX2[sb + 8 * d1];
#pragma unroll
                for (int k = 0; k < 8; ++k) {
                    v2f a;
                    a.x = ap[4 * k];
                    a.y = ap[4 * k + 1];
                    v2f bf;   // shared by both d-tiles (mirror symmetry)
                    bf.x = b0p[4 * k]     + b1p[4 * k];
                    bf.y = b0p[4 * k + 1] + b1p[4 * k + 1];
                    c0 = __builtin_amdgcn_wmma_f32_16x16x4_f32(
                            false, a, false, bf, (short)0, c0, false, false);
                    c1 = __builtin_amdgcn_wmma_f32_16x16x4_f32(
                            false, a, false, bf, (short)0, c1, false, false);
                }
            }
        }
    } else {
        // d = 0 and d = 8 are self-paired: B = 2*X[p]
        for (int dy = 0; dy < 3; ++dy) {
            for (int dx = 0; dx < 3; ++dx) {
                const float* ap  = &sW2[(dy * 3 + dx) * (16 * QP) + col * QP + 2 * hi];
                const int    sb  = (dy * 18 + col + dx) * CP + 2 + 2 * hi;
                const float* b0p = &sX2[sb + 8 * 0];
                const float* b1p = &sX2[sb + 8 * 8];
#pragma unroll
                for (int k = 0; k < 8; ++k) {
                    v2f a;
                    a.x = ap[4 * k];
                    a.y = ap[4 * k + 1];
                    float u0 = b0p[4 * k];
                    float u1 = b0p[4 * k + 1];
                    float v0 = b1p[4 * k];
                    float v1 = b1p[4 * k + 1];
                    v2f bf0, bf1;
                    bf0.x = u0 + u0;  bf0.y = u1 + u1;
                    bf1.x = v0 + v0;  bf1.y = v1 + v1;
                    c0 = __builtin_amdgcn_wmma_f32_16x16x4_f32(
                            false, a, false, bf0, (short)0, c0, false, false);
                    c1 = __builtin_amdgcn_wmma_f32_16x16x4_f32(
                            false, a, false, bf1, (short)0, c1, false, false);
                }
            }
        }
    }

    // --- store: C/D layout: VGPR r -> filter n = r + 8*hi, pixel = col ---
    float* op0 = out + ((b * COUT + d0 * 16 + hi * 8) * LL + y) * LL + x0 + col;
    float* op1 = out + ((b * COUT + d1 * 16 + hi * 8) * LL + y) * LL + x0 + col;
#pragma unroll
    for (int r = 0; r < 8; ++r) {
        op0[r * (LL * LL)] = c0[r];
        op1[r * (LL * LL)] = c1[r];
    }
}

extern "C" void kernel_launch(void* const* d_in, const int* in_sizes, int n_in,
                              void* d_out, int out_size, void* d_ws, size_t ws_size,
                              hipStream_t stream) {
    const float* x = (const float*)d_in[0];   // (16,128,32,32) f32
    const float* w = (const float*)d_in[1];   // (16,32,3,3)    f32
    float* out = (float*)d_out;               // (16,256,32,32) f32

    dim3 grid(BB * LL * 2);   // (b, y, x-half)
    dim3 block(256);
    hipLaunchKernelGGL(fconv2d_wmma_kernel, grid, block, 0, stream, x, w, out);
}